// AGNNConv_37031208026112
// MI455X (gfx1250) — compile-verified
//
#include <hip/hip_runtime.h>

typedef __attribute__((ext_vector_type(2))) float v2f;
typedef __attribute__((ext_vector_type(4))) float v4f;
typedef __attribute__((ext_vector_type(8))) float v8f;

#define N_NODES 50000
#define N_EDGES 800000
#define D_FEAT  64
#define E_TOT   (N_EDGES + N_NODES)          // 850000 edges incl. self loops
#define GROUPS  (E_TOT / 16)                 // 53125, exact (850000 = 53125*16)

// Workspace layout (float/uint elements):
//   [0, E_TOT)                     : w  (edge logits, then exp values)
//   [E_TOT, E_TOT+N)               : invn (1/norm per node)
//   [E_TOT+N, E_TOT+2N)            : wmax keys (order-preserving uint encoding)
//   [E_TOT+2N, E_TOT+3N)           : denom (softmax denominators)
#define WS_W      0
#define WS_INVN   (E_TOT)
#define WS_WMAX   (E_TOT + N_NODES)
#define WS_DENOM  (E_TOT + 2 * N_NODES)

// ---------------------------------------------------------------------------
// Kernel 0: zero out[], denom[], wmax keys (key 0 == identity for max encode)
// ---------------------------------------------------------------------------
__global__ __launch_bounds__(256) void init_kernel(float* __restrict__ out,
                                                   unsigned* __restrict__ wmaxkey,
                                                   float* __restrict__ denom) {
  int i = blockIdx.x * blockDim.x + threadIdx.x;
  if (i < N_NODES * D_FEAT) out[i] = 0.0f;
  if (i < N_NODES) { wmaxkey[i] = 0u; denom[i] = 0.0f; }
}

// ---------------------------------------------------------------------------
// Kernel 1: per-node inverse norm. One wave32 per node; 2 floats per lane.
// ---------------------------------------------------------------------------
__global__ __launch_bounds__(256) void invnorm_kernel(const float* __restrict__ x,
                                                      float* __restrict__ invn) {
  int gid  = blockIdx.x * blockDim.x + threadIdx.x;
  int node = gid >> 5;
  int lane = threadIdx.x & 31;
  if (node >= N_NODES) return;
  const float* xr = x + (long)node * D_FEAT;
  float a = xr[lane];
  float b = xr[lane + 32];
  float s = a * a + b * b;
#pragma unroll
  for (int m = 16; m >= 1; m >>= 1) s += __shfl_xor(s, m, 32);
  if (lane == 0) invn[node] = 1.0f / fmaxf(sqrtf(s), 1e-12f);
}

// ---------------------------------------------------------------------------
// Kernel 2: edge logits via fp32 WMMA. One wave handles 16 edges.
//   A[m][k] = x[row[e_m]][k0+k]  (16x4, lane m<16 holds K=0,1; lane m+16 K=2,3)
//   B[k][n] = x[col[e_n]][k0+k]  (4x16, same per-lane pattern, source rows)
//   D = sum_chunks A*B ; diagonal D[e][e] = dot(x[row_e], x[col_e])
// ---------------------------------------------------------------------------
__global__ __launch_bounds__(256) void edge_w_kernel(const float* __restrict__ x,
                                                     const int* __restrict__ row,
                                                     const int* __restrict__ col,
                                                     const float* __restrict__ beta,
                                                     const float* __restrict__ invn,
                                                     float* __restrict__ w,
                                                     unsigned* __restrict__ wmaxkey) {
  int gid  = blockIdx.x * blockDim.x + threadIdx.x;
  int wave = gid >> 5;
  int lane = threadIdx.x & 31;
  if (wave >= GROUPS) return;                 // wave-uniform exit: EXEC stays full
  long ebase = (long)wave * 16;
  int  m     = lane & 15;                     // which edge of the group this lane feeds
  long eid   = ebase + m;
  int r = (eid < N_EDGES) ? row[eid] : (int)(eid - N_EDGES);   // self loop
  int c = (eid < N_EDGES) ? col[eid] : (int)(eid - N_EDGES);
  int koff = (lane >> 4) << 1;                // 0 for lanes 0-15, 2 for lanes 16-31
  const float* ar = x + (long)r * D_FEAT + koff;
  const float* bc = x + (long)c * D_FEAT + koff;

  v8f acc = {};
#pragma unroll
  for (int k0 = 0; k0 < D_FEAT; k0 += 4) {
    v2f a = *(const v2f*)(ar + k0);           // 8B-aligned global_load_b64
    v2f b = *(const v2f*)(bc + k0);
    acc = __builtin_amdgcn_wmma_f32_16x16x4_f32(
        /*neg_a=*/false, a, /*neg_b=*/false, b,
        /*c_mod=*/(short)0, acc, /*reuse_a=*/false, /*reuse_b=*/false);
  }

  // Diagonal extraction per documented 16x16 f32 C/D layout:
  //   e<8  -> lane e,     vgpr e        ;  e>=8 -> lane e+16, vgpr e-8
  float d = 0.0f;
#pragma unroll
  for (int i = 0; i < 8; ++i) d += ((lane & 7) == i) ? acc[i] : 0.0f;

  bool active = (lane < 8) || (lane >= 24);   // these lanes hold diag for edge (lane&15)
  if (active) {
    float wv = beta[0] * invn[r] * invn[c] * d;
    w[eid] = wv;
    unsigned bits = __float_as_uint(wv);
    unsigned key  = (bits & 0x80000000u) ? ~bits : (bits | 0x80000000u);
    atomicMax(&wmaxkey[r], key);              // segment max, order-preserving encode
  }
}

// ---------------------------------------------------------------------------
// Kernel 3: e = exp(w - wmax[row]); denom[row] += e. One thread per edge.
// ---------------------------------------------------------------------------
__global__ __launch_bounds__(256) void edge_exp_kernel(const int* __restrict__ row,
                                                       float* __restrict__ w,
                                                       const unsigned* __restrict__ wmaxkey,
                                                       float* __restrict__ denom) {
  int e = blockIdx.x * blockDim.x + threadIdx.x;
  if (e >= E_TOT) return;
  int r = (e < N_EDGES) ? row[e] : e - N_EDGES;
  unsigned key = wmaxkey[r];
  unsigned mb  = (key & 0x80000000u) ? (key & 0x7FFFFFFFu) : ~key;
  float ev = __expf(w[e] - __uint_as_float(mb));
  w[e] = ev;
  atomicAdd(&denom[r], ev);
}

// ---------------------------------------------------------------------------
// Kernel 4: out[row] += (e/denom[row]) * x[col].  16 lanes per edge, float4
// per lane -> contiguous 256B gather, 4x global_atomic_add_f32 per lane.
// ---------------------------------------------------------------------------
__global__ __launch_bounds__(256) void aggregate_kernel(const float* __restrict__ x,
                                                        const int* __restrict__ row,
                                                        const int* __restrict__ col,
                                                        const float* __restrict__ w,
                                                        const float* __restrict__ denom,
                                                        float* __restrict__ out) {
  long t = (long)blockIdx.x * blockDim.x + threadIdx.x;
  long e = t >> 4;
  if (e >= E_TOT) return;
  int f = (int)(t & 15) * 4;
  int r = (e < N_EDGES) ? row[e] : (int)(e - N_EDGES);
  int c = (e < N_EDGES) ? col[e] : (int)(e - N_EDGES);
  float alpha = w[e] / (denom[r] + 1e-16f);
  v4f xv = *(const v4f*)(x + (long)c * D_FEAT + f);   // 16B-aligned b128 load
  float* o = out + (long)r * D_FEAT + f;
  atomicAdd(o + 0, alpha * xv.x);
  atomicAdd(o + 1, alpha * xv.y);
  atomicAdd(o + 2, alpha * xv.z);
  atomicAdd(o + 3, alpha * xv.w);
}

// ---------------------------------------------------------------------------
extern "C" void kernel_launch(void* const* d_in, const int* in_sizes, int n_in,
                              void* d_out, int out_size, void* d_ws, size_t ws_size,
                              hipStream_t stream) {
  const float* x    = (const float*)d_in[0];
  const int*   ei   = (const int*)d_in[1];        // [2, E] flat: row then col
  const float* beta = (const float*)d_in[2];
  const int* row = ei;
  const int* col = ei + N_EDGES;

  float*    ws      = (float*)d_ws;
  float*    w       = ws + WS_W;
  float*    invn    = ws + WS_INVN;
  unsigned* wmaxkey = (unsigned*)(ws + WS_WMAX);
  float*    denom   = ws + WS_DENOM;
  float*    out     = (float*)d_out;

  // 0) init accumulators (required every call: atomics + poisoned buffers)
  {
    int n = N_NODES * D_FEAT;
    init_kernel<<<(n + 255) / 256, 256, 0, stream>>>(out, wmaxkey, denom);
  }
  // 1) per-node inverse norms: one wave per node
  {
    long threads = (long)N_NODES * 32;
    invnorm_kernel<<<(int)((threads + 255) / 256), 256, 0, stream>>>(x, invn);
  }
  // 2) edge logits via fp32 WMMA: one wave per 16 edges (exact tiling)
  {
    long threads = (long)GROUPS * 32;
    edge_w_kernel<<<(int)((threads + 255) / 256), 256, 0, stream>>>(
        x, row, col, beta, invn, w, wmaxkey);
  }
  // 3) segment softmax numerator + denominator
  edge_exp_kernel<<<(E_TOT + 255) / 256, 256, 0, stream>>>(row, w, wmaxkey, denom);
  // 4) weighted aggregation (scatter-add)
  {
    long threads = (long)E_TOT * 16;
    aggregate_kernel<<<(int)((threads + 255) / 256), 256, 0, stream>>>(
        x, row, col, w, denom, out);
  }
}